// data_consistency_fista_83734682403051
// MI455X (gfx1250) — compile-verified
//
#include <hip/hip_runtime.h>
#include <math.h>

// Problem constants (B=4, C=16, H=W=384, 20 FISTA iterations)
#define NN     384
#define PL     (384 * 384)        // 147456 floats per plane
#define NB     4
#define NC     16
#define NITER  20
#define KBLK   16
#define MACM   32                 // macro-tile M
#define MACN   64                 // macro-tile N
#define ASTR   18                 // As row stride (floats): 8B-aligned, bank-spread
#define BSTR   68                 // Bs row stride (floats): 16B-aligned, bank-spread

typedef __attribute__((ext_vector_type(2))) float v2f;
typedef __attribute__((ext_vector_type(8))) float v8f;
typedef __attribute__((ext_vector_type(2))) int   i2v;
typedef __attribute__((ext_vector_type(4))) int   i4v;

#ifndef __has_builtin
#define __has_builtin(x) 0
#endif

// ---- CDNA5 async global->LDS path (ASYNCcnt), with graceful fallback ------
#if __has_builtin(__builtin_amdgcn_global_load_async_to_lds_b64) && \
    __has_builtin(__builtin_amdgcn_global_load_async_to_lds_b128)
#define HAVE_ASYNC 1
#else
#define HAVE_ASYNC 0
#endif

// Builtin params are (global vNi* src, local vNi* dst, imm offset, imm cpol);
// pointee is non-const, so strip const before the address-space cast.
#define GPTR64(p)  ((__attribute__((address_space(1))) i2v*)(float*)(p))
#define LPTR64(p)  ((__attribute__((address_space(3))) i2v*)(p))
#define GPTR128(p) ((__attribute__((address_space(1))) i4v*)(float*)(p))
#define LPTR128(p) ((__attribute__((address_space(3))) i4v*)(p))

#if HAVE_ASYNC
#if __has_builtin(__builtin_amdgcn_s_wait_asynccnt)
#define WAIT_ASYNC(n) __builtin_amdgcn_s_wait_asynccnt(n)
#else
#define WAIT_ASYNC(n) asm volatile("s_wait_asynccnt " #n ::: "memory")
#endif
#else
#define WAIT_ASYNC(n) do {} while (0)
#endif

static __device__ __forceinline__ v8f wmma_f32(v2f a, v2f b, v8f c) {
    // V_WMMA_F32_16X16X4_F32 : full fp32 matrix core op (CDNA5)
    return __builtin_amdgcn_wmma_f32_16x16x4_f32(
        /*neg_a=*/false, a, /*neg_b=*/false, b,
        /*c_mod=*/(short)0, c, /*reuse_a=*/false, /*reuse_b=*/false);
}

// ---------------------------------------------------------------------------
// Complex 384x384x384 GEMM, C = A * B (complex, SoA re/im planes), with
// stage-specific fused elementwise ops.
//   STAGE 1: A = coil*y (computed on load), B = F      -> BUF1 (R = G*F)
//   STAGE 2: A = F, B = BUF1; post: mask*(out - x0)    -> BUF2 (K)
//   STAGE 3: A = BUF2, B = Fi                          -> BUF1 (R2 = K*Fi)
//   STAGE 4: A = Fi, B = BUF1; post: conj(coil)*out    -> BUF2 (per-coil grad)
// Grid: (64 pairs, 12, 6), block 256 (8 wave32 waves, 32x64 macro-tile).
// Double-buffered LDS K-blocks filled by GLOBAL_LOAD_ASYNC_TO_LDS.
// ---------------------------------------------------------------------------
template <int STAGE>
__global__ __launch_bounds__(256)
void cgemm384(const float* __restrict__ Are_g, const float* __restrict__ Aim_g,
              const float* __restrict__ Bre_g, const float* __restrict__ Bim_g,
              float* __restrict__ Cbase,
              const float* __restrict__ csm, const float* __restrict__ yimg,
              const float* __restrict__ x0c, const int* __restrict__ mask)
{
    __shared__ __align__(16) float As[2][2][MACM][ASTR];   // 9,216 B
    __shared__ __align__(16) float Bs[2][2][KBLK][BSTR];   // 17,408 B

    const int p  = blockIdx.x;            // (batch, coil) pair
    const int b  = p >> 4;
    const int c  = p & 15;
    const int M0 = blockIdx.y * MACM;
    const int N0 = blockIdx.z * MACN;
    const int tid = threadIdx.x;

    constexpr bool A_PAIR = (STAGE == 3);
    constexpr bool B_PAIR = (STAGE == 2 || STAGE == 4);
    const size_t poff = (size_t)p * 2 * PL;

    const float* Are = A_PAIR ? (Are_g + poff)      : Are_g;
    const float* Aim = A_PAIR ? (Are_g + poff + PL) : Aim_g;
    const float* Bre = B_PAIR ? (Bre_g + poff)      : Bre_g;
    const float* Bim = B_PAIR ? (Bre_g + poff + PL) : Bim_g;
    float* Cre = Cbase + poff;
    float* Cim = Cre + PL;

    // Stage-1 / stage-4 coil & y plane pointers
    const float* cre_p = nullptr; const float* cim_p = nullptr;
    const float* yre_p = nullptr; const float* yim_p = nullptr;
    if (STAGE == 1 || STAGE == 4) {
        cre_p = csm + ((size_t)(b * 2 + 0) * NC + c) * PL;
        cim_p = csm + ((size_t)(b * 2 + 1) * NC + c) * PL;
    }
    if (STAGE == 1) {
        yre_p = yimg + (size_t)(b * 2 + 0) * PL;
        yim_p = yimg + (size_t)(b * 2 + 1) * PL;
    }

    // Staging split: A-block 32x16 (8 thr/row x 2 floats), B-block 16x64 (x4)
    const int arow = tid >> 3;             // 0..31
    const int acol = (tid & 7) << 1;       // 0,2,..,14  (8B-aligned)
    const int brow = tid >> 4;             // 0..15
    const int bcol = (tid & 15) << 2;      // 0,4,..,60  (16B-aligned)

    // Wave / lane decomposition (wave32): 8 waves = 2(M) x 4(N) tiles
    const int l    = tid & 31;
    const int lr   = l & 15;
    const int half = l >> 4;
    const int wid  = tid >> 5;             // 0..7
    const int moff = (wid & 1) * 16;       // 0 or 16
    const int noff = (wid >> 1) * 16;      // 0,16,32,48

    v8f accR = {};
    v8f accI = {};

    auto stage_block = [&](int buf, int kk) {
        // ---- A block ----
        if (STAGE == 1) {
            // G = coil .* y computed in VALU, stored to LDS (b64 store)
            #pragma unroll
            for (int j = 0; j < 2; ++j) {
                const size_t g = (size_t)(M0 + arow) * NN + (kk + acol + j);
                const float cr = cre_p[g], ci = cim_p[g];
                const float yr = yre_p[g], yi = yim_p[g];
                As[buf][0][arow][acol + j] = cr * yr - ci * yi;
                As[buf][1][arow][acol + j] = cr * yi + ci * yr;
            }
        } else {
            const size_t g = (size_t)(M0 + arow) * NN + (kk + acol);
#if HAVE_ASYNC
            __builtin_amdgcn_global_load_async_to_lds_b64(
                GPTR64(Are + g), LPTR64(&As[buf][0][arow][acol]), 0, 0);
            __builtin_amdgcn_global_load_async_to_lds_b64(
                GPTR64(Aim + g), LPTR64(&As[buf][1][arow][acol]), 0, 0);
#else
            #pragma unroll
            for (int j = 0; j < 2; ++j) {
                As[buf][0][arow][acol + j] = Are[g + j];
                As[buf][1][arow][acol + j] = Aim[g + j];
            }
#endif
        }
        // ---- B block ----
        {
            const size_t g = (size_t)(kk + brow) * NN + (N0 + bcol);
#if HAVE_ASYNC
            __builtin_amdgcn_global_load_async_to_lds_b128(
                GPTR128(Bre + g), LPTR128(&Bs[buf][0][brow][bcol]), 0, 0);
            __builtin_amdgcn_global_load_async_to_lds_b128(
                GPTR128(Bim + g), LPTR128(&Bs[buf][1][brow][bcol]), 0, 0);
#else
            #pragma unroll
            for (int j = 0; j < 4; ++j) {
                Bs[buf][0][brow][bcol + j] = Bre[g + j];
                Bs[buf][1][brow][bcol + j] = Bim[g + j];
            }
#endif
        }
    };

    // ---- double-buffered K loop: async-fill block k+1 while WMMA eats k ----
    stage_block(0, 0);
    int buf = 0;
    for (int kk = 0; kk < NN; kk += KBLK) {
        const int nxt = kk + KBLK;
        if (nxt < NN) stage_block(buf ^ 1, nxt);

        if (nxt < NN) {
            if (STAGE == 1) { WAIT_ASYNC(2); }   // only B in flight for next blk
            else           { WAIT_ASYNC(4); }   // A(2 x b64) + B(2 x b128)
        } else {
            WAIT_ASYNC(0);
        }
        __syncthreads();

        const int mrow = moff + lr;
        const int nc   = noff + lr;
        #pragma unroll
        for (int k2 = 0; k2 < KBLK; k2 += 4) {
            const int k0 = k2 + 2 * half;
            v2f ar, ai;
            ar.x = As[buf][0][mrow][k0];
            ar.y = As[buf][0][mrow][k0 + 1];
            ai.x = As[buf][1][mrow][k0];
            ai.y = As[buf][1][mrow][k0 + 1];
            const v2f an = -ai;   // f32 WMMA has no A-neg modifier: VALU negate

            v2f br, bi;
            br.x = Bs[buf][0][k0][nc];
            br.y = Bs[buf][0][k0 + 1][nc];
            bi.x = Bs[buf][1][k0][nc];
            bi.y = Bs[buf][1][k0 + 1][nc];

            accR = wmma_f32(ar, br, accR);   // + Are*Bre
            accR = wmma_f32(an, bi, accR);   // - Aim*Bim
            accI = wmma_f32(ar, bi, accI);   // + Are*Bim
            accI = wmma_f32(ai, br, accI);   // + Aim*Bre
        }
        __syncthreads();
        buf ^= 1;
    }

    // ---- epilogue with fused elementwise ops ----
    #pragma unroll
    for (int e = 0; e < 8; ++e) {
        const int row = M0 + moff + e + 8 * half;   // C-layout: M = e + 8*half
        const int col = N0 + noff + lr;             //           N = lane&15
        const size_t idx = (size_t)row * NN + col;
        float tre = accR[e];
        float tim = accI[e];
        if (STAGE == 2) {
            const float mv = (float)mask[(size_t)b * PL + idx];
            tre = mv * (tre - x0c[(size_t)(b * 2 + 0) * PL + idx]);
            tim = mv * (tim - x0c[(size_t)(b * 2 + 1) * PL + idx]);
        }
        if (STAGE == 4) {                 // conj(coil) * out
            const float cr = cre_p[idx], ci = cim_p[idx];
            const float r = cr * tre + ci * tim;
            const float i = cr * tim - ci * tre;
            tre = r; tim = i;
        }
        Cre[idx] = tre;
        Cim[idx] = tim;
    }
}

// ---------------------------------------------------------------------------
// DFT matrices: F = exp(-2*pi*i*jk/N), Fi = exp(+2*pi*i*jk/N)/N  (1/N per pass)
// ---------------------------------------------------------------------------
__global__ void fill_dft(float* __restrict__ Fre, float* __restrict__ Fim,
                         float* __restrict__ Fire, float* __restrict__ Fiim)
{
    const int idx = blockIdx.x * 256 + threadIdx.x;   // < PL
    const int j = idx / NN, k = idx % NN;
    const int t = (j * k) % NN;                       // exact arg reduction
    const float ph = (float)t * (6.283185307179586477f / (float)NN);
    const float cs = cosf(ph), sn = sinf(ph);
    Fre[idx]  = cs;
    Fim[idx]  = -sn;
    Fire[idx] = cs * (1.0f / (float)NN);
    Fiim[idx] = sn * (1.0f / (float)NN);
}

__global__ void zero_kernel(float* __restrict__ p, size_t n)
{
    const size_t i = (size_t)blockIdx.x * 256 + threadIdx.x;
    if (i < n) p[i] = 0.0f;
}

// grad[b][ri][i] = sum_c BUF2[(b*16+c)][ri][i]
__global__ void reduce_coils(const float* __restrict__ buf, float* __restrict__ grad)
{
    const size_t idx = (size_t)blockIdx.x * 256 + threadIdx.x;   // < B*2*PL
    const size_t b = idx / (2 * (size_t)PL);
    const size_t rem = idx % (2 * (size_t)PL);
    const float* src = buf + b * NC * 2 * (size_t)PL + rem;
    float s = 0.0f;
    #pragma unroll
    for (int c = 0; c < NC; ++c) s += src[(size_t)c * 2 * PL];
    grad[idx] = s;
}

// FISTA pointwise: xn = prox((1-lam)*(y-grad) + lam*z); y = xn + coef*(xn-x)
__global__ void fista_update(const float* __restrict__ grad,
                             const float* __restrict__ zk,
                             const float* __restrict__ lam,
                             float* __restrict__ X, float* __restrict__ Y,
                             float coef)
{
    const size_t idx = (size_t)blockIdx.x * 256 + threadIdx.x;   // < B*PL
    const size_t b = idx / PL, i = idx % PL;
    const size_t re = (b * 2) * (size_t)PL + i, im = re + PL;
    const float lm = lam[0];
    float xr = Y[re] - grad[re];
    float xi = Y[im] - grad[im];
    xr = xr - lm * (xr - zk[re]);
    xi = xi - lm * (xi - zk[im]);
    xr = fmaxf(xr, 0.0f);
    xi = fmaxf(xi, 0.0f);
    const float yr = xr + coef * (xr - X[re]);
    const float yi = xi + coef * (xi - X[im]);
    X[re] = xr; X[im] = xi;
    Y[re] = yr; Y[im] = yi;
}

__global__ void copy_out(const float* __restrict__ src, float* __restrict__ dst)
{
    const size_t i = (size_t)blockIdx.x * 256 + threadIdx.x;     // < B*2*PL
    dst[i] = src[i];
}

extern "C" void kernel_launch(void* const* d_in, const int* in_sizes, int n_in,
                              void* d_out, int out_size, void* d_ws, size_t ws_size,
                              hipStream_t stream)
{
    const float* zk  = (const float*)d_in[0];   // [B,2,H,W]
    const float* x0  = (const float*)d_in[1];   // [B,2,H,W] (measured data b)
    const float* csm = (const float*)d_in[2];   // [B,2,C,H,W]
    const float* lam = (const float*)d_in[3];   // [1]
    const int*   msk = (const int*)d_in[4];     // [B,H,W]
    (void)in_sizes; (void)n_in; (void)out_size; (void)ws_size;

    float* ws   = (float*)d_ws;
    float* Fre  = ws;
    float* Fim  = Fre  + (size_t)PL;
    float* Fire = Fim  + (size_t)PL;
    float* Fiim = Fire + (size_t)PL;
    float* X    = Fiim + (size_t)PL;            // [B][2][PL]
    float* Y    = X    + 8 * (size_t)PL;        // [B][2][PL]
    float* GRAD = Y    + 8 * (size_t)PL;        // [B][2][PL]
    float* BUF1 = GRAD + 8 * (size_t)PL;        // [64][2][PL]  (~75.5 MB)
    float* BUF2 = BUF1 + 128 * (size_t)PL;      // [64][2][PL]  (~75.5 MB)

    fill_dft<<<PL / 256, 256, 0, stream>>>(Fre, Fim, Fire, Fiim);
    zero_kernel<<<(16 * PL) / 256, 256, 0, stream>>>(X, 16 * (size_t)PL); // X and Y

    const dim3 grid(NB * NC, NN / MACM, NN / MACN);   // (64, 12, 6)
    double t = 1.0;
    for (int it = 0; it < NITER; ++it) {
        const double tn = (1.0 + sqrt(1.0 + 4.0 * t * t)) * 0.5;
        const float coef = (float)((t - 1.0) / tn);     // data-independent momentum
        t = tn;

        // R  = (coil .* y) * F          (row DFT)
        cgemm384<1><<<grid, 256, 0, stream>>>(nullptr, nullptr, Fre, Fim,
                                              BUF1, csm, Y, nullptr, nullptr);
        // K  = mask .* (F * R - b)      (col DFT + data consistency)
        cgemm384<2><<<grid, 256, 0, stream>>>(Fre, Fim, BUF1, nullptr,
                                              BUF2, nullptr, nullptr, x0, msk);
        // R2 = K * Fi                   (row IDFT)
        cgemm384<3><<<grid, 256, 0, stream>>>(BUF2, nullptr, Fire, Fiim,
                                              BUF1, nullptr, nullptr, nullptr, nullptr);
        // Gc = conj(coil) .* (Fi * R2)  (col IDFT + coil adjoint)
        cgemm384<4><<<grid, 256, 0, stream>>>(Fire, Fiim, BUF1, nullptr,
                                              BUF2, csm, nullptr, nullptr, nullptr);
        // grad = sum_c Gc
        reduce_coils<<<(8 * PL) / 256, 256, 0, stream>>>(BUF2, GRAD);
        // FISTA pointwise update
        fista_update<<<(4 * PL) / 256, 256, 0, stream>>>(GRAD, zk, lam, X, Y, coef);
    }
    copy_out<<<(8 * PL) / 256, 256, 0, stream>>>(X, (float*)d_out);
}